// SeKG_Module_66838281060868
// MI455X (gfx1250) — compile-verified
//
#include <hip/hip_runtime.h>
#include <stdint.h>

typedef __attribute__((ext_vector_type(2))) float v2f;
typedef __attribute__((ext_vector_type(8))) float v8f;

#define NB 5            // batches actually used by the gather
#define CD 256          // channels
#define WIN_ELEMS (NB*CD*9)   // 11520 floats = 46080 B

// ---------------------------------------------------------------------------
// Kernel 1: one block. pooled -> channel convs -> FC1 (WMMA f32) -> FC2
// (WMMA f32) -> sigmoid -> win[k][c][3][3] = x[k,c,3:6,3:6] * scale[k][c]
// ---------------------------------------------------------------------------
__global__ __launch_bounds__(256) void sekg_prep(
    const float* __restrict__ x,
    const float* __restrict__ w_c1, const float* __restrict__ b_c1,
    const float* __restrict__ w_c2, const float* __restrict__ b_c2,
    const float* __restrict__ w_c3, const float* __restrict__ b_c3,
    const float* __restrict__ w_fc1, const float* __restrict__ b_fc1,
    const float* __restrict__ w_fc2, const float* __restrict__ b_fc2,
    float* __restrict__ win)
{
  __shared__ float pooled[NB*CD];
  // att padded to 16 "batch" rows (rows 5..15 are zero) so the WMMA
  // B-fragment load is an unconditional ds_load_b64 -- no exec masking.
  __shared__ alignas(16) float att[16*CD];
  __shared__ alignas(16) float h_lds[16*16];   // h[o][n] (o = FC2 K, n = batch)
  __shared__ float scale_lds[NB*CD];

  const int tid  = threadIdx.x;
  const int lane = tid & 31;
  const int wave = tid >> 5;
  const int hlf  = lane >> 4;   // 0: lanes 0-15, 1: lanes 16-31
  const int mn   = lane & 15;   // M (A rows) == N (B cols) lane index

  // ---- Phase A: pooled[k*256+c] = mean over 81 pixels (wave-cooperative) --
  for (int p = wave; p < NB*CD; p += 8) {
    const float* src = x + p * 81;                 // (k*256+c)*81 contiguous
    float v = src[lane] + src[lane + 32] + ((lane < 17) ? src[lane + 64] : 0.0f);
    #pragma unroll
    for (int off = 16; off > 0; off >>= 1) v += __shfl_xor(v, off, 32);
    if (lane == 0) pooled[p] = v * (1.0f / 81.0f);
  }
  __syncthreads();

  // ---- Phase B: 1-D channel convs (middle rows of the 3/5/7 kernels) -----
  {
    float t3[3], t5[5], t7[7];
    #pragma unroll
    for (int i = 0; i < 3; ++i) t3[i] = w_c1[3 + i];   // row 1 of 3x3
    #pragma unroll
    for (int i = 0; i < 5; ++i) t5[i] = w_c2[10 + i];  // row 2 of 5x5
    #pragma unroll
    for (int i = 0; i < 7; ++i) t7[i] = w_c3[21 + i];  // row 3 of 7x7
    const float bsum = b_c1[0] + b_c2[0] + b_c3[0];
    for (int p = tid; p < 16*CD; p += 256) {
      const int k = p >> 8, c = p & 255;
      if (k < NB) {
        const float* P = pooled + k * CD;
        float acc = P[c] + bsum;
        #pragma unroll
        for (int t = 0; t < 3; ++t) { int cc = c - 1 + t; if (cc >= 0 && cc < CD) acc += t3[t] * P[cc]; }
        #pragma unroll
        for (int t = 0; t < 5; ++t) { int cc = c - 2 + t; if (cc >= 0 && cc < CD) acc += t5[t] * P[cc]; }
        #pragma unroll
        for (int t = 0; t < 7; ++t) { int cc = c - 3 + t; if (cc >= 0 && cc < CD) acc += t7[t] * P[cc]; }
        att[p] = acc;
      } else {
        att[p] = 0.0f;                 // zero pad: batch cols 5..15
      }
    }
  }
  __syncthreads();

  // ---- Phase C: FC1 = relu(W1(16x256) * att(256x16) + b1), wave 0 only ---
  // V_WMMA_F32_16X16X4_F32, K accumulated in 64 steps of 4.
  if (wave == 0) {
    v8f acc = {};
    for (int ks = 0; ks < 64; ++ks) {
      const int k0 = ks * 4 + hlf * 2;
      const v2f a = *(const v2f*)&w_fc1[mn * CD + k0];  // A[m][k0..k0+1]
      const v2f b = *(const v2f*)&att[mn * CD + k0];    // B[k0..k0+1][n]
      acc = __builtin_amdgcn_wmma_f32_16x16x4_f32(
          false, a, false, b, (short)0, acc, false, false);
    }
    #pragma unroll
    for (int r = 0; r < 8; ++r) {       // D: VGPR r -> M = r + 8*hlf, N = mn
      const int o = hlf * 8 + r;
      float hv = acc[r] + b_fc1[o];
      h_lds[o * 16 + mn] = hv > 0.0f ? hv : 0.0f;
    }
  }
  __syncthreads();

  // ---- Phase D: FC2 = sigmoid(W2(256x16) * h(16x16) + b2), 8 waves -------
  #pragma unroll
  for (int t = 0; t < 2; ++t) {
    const int mt = wave * 2 + t;        // M-tile: rows mt*16 .. mt*16+15
    v8f acc = {};
    #pragma unroll
    for (int ks = 0; ks < 4; ++ks) {
      const int k0 = ks * 4 + hlf * 2;
      const v2f a = *(const v2f*)&w_fc2[(mt * 16 + mn) * 16 + k0];
      v2f b;
      b.x = h_lds[k0 * 16 + mn];
      b.y = h_lds[(k0 + 1) * 16 + mn];
      acc = __builtin_amdgcn_wmma_f32_16x16x4_f32(
          false, a, false, b, (short)0, acc, false, false);
    }
    #pragma unroll
    for (int r = 0; r < 8; ++r) {
      const int co = mt * 16 + hlf * 8 + r;
      const float z = acc[r] + b_fc2[co];
      const float s = 1.0f / (1.0f + __expf(-z));
      if (mn < NB) scale_lds[mn * CD + co] = s;
    }
  }
  __syncthreads();

  // ---- Phase E: win[k][c][i][j] = x[k,c,3+i,3+j] * scale[k][c] -----------
  for (int w = tid; w < WIN_ELEMS; w += 256) {
    const int k = w / (CD * 9);
    const int rem = w - k * (CD * 9);
    const int c = rem / 9;
    const int q = rem - c * 9;
    const int i2 = q / 3, j2 = q - i2 * 3;
    const float xv = x[(k * CD + c) * 81 + (3 + i2) * 9 + (3 + j2)];
    win[w] = xv * scale_lds[k * CD + c];
  }
}

// ---------------------------------------------------------------------------
// Kernel 2: gather. Stage win table (46 KB) + this batch's idx row (2 KB)
// into LDS with async-to-LDS b128 copies, then stream coalesced output:
//   out[b,k,j,i2,j2] = win[4-k][ idx[b,4-k,j] ][i2][j2]
// ---------------------------------------------------------------------------
__global__ __launch_bounds__(256) void sekg_gather(
    const float* __restrict__ win, const int* __restrict__ idx,
    float* __restrict__ out)
{
  __shared__ alignas(16) float win_lds[WIN_ELEMS];   // 46080 B
  __shared__ alignas(16) int   idx_lds[500];         //  2000 B
  const int tid = threadIdx.x;
  const int b   = blockIdx.x;

  const unsigned win_base = (unsigned)(uintptr_t)&win_lds[0];
  const unsigned idx_base = (unsigned)(uintptr_t)&idx_lds[0];

  // async stage win: 2880 x 16B chunks (tracked by ASYNCcnt)
  for (int m = tid; m < WIN_ELEMS / 4; m += 256) {
    const unsigned loff = win_base + (unsigned)m * 16u;
    const unsigned long long g = (unsigned long long)(uintptr_t)(win + m * 4);
    asm volatile("global_load_async_to_lds_b128 %0, %1, off"
                 :: "v"(loff), "v"(g) : "memory");
  }
  // async stage idx row: 125 x 16B chunks
  if (tid < 125) {
    const unsigned loff = idx_base + (unsigned)tid * 16u;
    const unsigned long long g =
        (unsigned long long)(uintptr_t)(idx + (size_t)b * 500 + tid * 4);
    asm volatile("global_load_async_to_lds_b128 %0, %1, off"
                 :: "v"(loff), "v"(g) : "memory");
  }
  asm volatile("s_wait_asynccnt 0x0" ::: "memory");
  __syncthreads();

  float* o = out + (size_t)b * 4500;
  for (int e = tid; e < 4500; e += 256) {
    const int k  = e / 900;
    const int r  = e - k * 900;
    const int j  = r / 9;
    const int q  = r - j * 9;
    const int kk = 4 - k;
    const int c  = idx_lds[kk * 100 + j];
    o[e] = win_lds[(kk * CD + c) * 9 + q];
  }
}

// ---------------------------------------------------------------------------
extern "C" void kernel_launch(void* const* d_in, const int* in_sizes, int n_in,
                              void* d_out, int out_size, void* d_ws, size_t ws_size,
                              hipStream_t stream) {
  const float* x     = (const float*)d_in[0];
  const float* w_c1  = (const float*)d_in[1];
  const float* b_c1  = (const float*)d_in[2];
  const float* w_c2  = (const float*)d_in[3];
  const float* b_c2  = (const float*)d_in[4];
  const float* w_c3  = (const float*)d_in[5];
  const float* b_c3  = (const float*)d_in[6];
  const float* w_fc1 = (const float*)d_in[7];
  const float* b_fc1 = (const float*)d_in[8];
  const float* w_fc2 = (const float*)d_in[9];
  const float* b_fc2 = (const float*)d_in[10];
  const int*   idx   = (const int*)d_in[11];
  float* win = (float*)d_ws;             // 46080 B scratch
  float* out = (float*)d_out;

  const int B = in_sizes[11] / 500;      // idx is (B,5,100)

  sekg_prep<<<1, 256, 0, stream>>>(x, w_c1, b_c1, w_c2, b_c2, w_c3, b_c3,
                                   w_fc1, b_fc1, w_fc2, b_fc2, win);
  sekg_gather<<<B, 256, 0, stream>>>(win, idx, out);
}